// EmeL_43825846288779
// MI455X (gfx1250) — compile-verified
//
#include <hip/hip_runtime.h>
#include <cstdint>
#include <cfloat>

typedef __attribute__((ext_vector_type(2))) float v2f;
typedef __attribute__((ext_vector_type(8))) float v8f;

#define NCHUNK 8

// ---------------------------------------------------------------------------
// Kernel 1: per-column partial sum & sum-of-squares via V_WMMA_F32_16X16X4_F32
//   C = ones(16x4) x Bchunk(4x16)  ->  every row of C holds the column sums.
//   A=ones makes the result invariant to the K->lane mapping of the B operand;
//   only the N striping (lane%16 -> column) matters.
// Each wave owns 16 consecutive columns; blockIdx.y picks a row chunk.
// ---------------------------------------------------------------------------
__global__ __launch_bounds__(256) void colstat_wmma(const float* __restrict__ h,
                                                    float* __restrict__ psum,
                                                    float* __restrict__ psumsq,
                                                    int Bdim, int D) {
  const int lane = threadIdx.x & 31;
  const int wave = threadIdx.x >> 5;
  const int half = lane >> 4;               // which 16-lane half of the wave
  const int col  = blockIdx.x * 128 + wave * 16 + (lane & 15);
  const int rowsPerChunk = Bdim / NCHUNK;
  const int rstart = blockIdx.y * rowsPerChunk;

  const size_t strideRow = (size_t)D;
  const float* pr = h + (size_t)(rstart + 2 * half) * strideRow + col;

  v2f a; a.x = 1.0f; a.y = 1.0f;            // A matrix: all ones
  v8f csum = {0.f,0.f,0.f,0.f,0.f,0.f,0.f,0.f};
  v8f csq  = {0.f,0.f,0.f,0.f,0.f,0.f,0.f,0.f};

  #pragma unroll 4
  for (int r = 0; r < rowsPerChunk; r += 4) {
    float b0 = pr[0];                       // row r + 2*half
    float b1 = pr[strideRow];               // row r + 2*half + 1
    v2f b;  b.x  = b0;      b.y  = b1;
    v2f bs; bs.x = b0 * b0; bs.y = b1 * b1;
    // 8 args: (neg_a, A, neg_b, B, c_mod, C, reuse_a, reuse_b)
    csum = __builtin_amdgcn_wmma_f32_16x16x4_f32(false, a, false, b,
                                                 (short)0, csum, false, false);
    csq  = __builtin_amdgcn_wmma_f32_16x16x4_f32(false, a, false, bs,
                                                 (short)0, csq,  false, false);
    pr += strideRow * 4;
  }

  // Row M=0 of C lives in VGPR0 of lanes 0..15 (N = lane). All rows identical.
  if (half == 0) {
    psum  [(size_t)blockIdx.y * D + col] = csum[0];
    psumsq[(size_t)blockIdx.y * D + col] = csq[0];
  }
}

// ---------------------------------------------------------------------------
// Kernel 2: fold the NCHUNK partial slabs, compute new_mean / new_var per
// column, and produce per-block partial sums of new_var (deterministic).
// ---------------------------------------------------------------------------
__global__ __launch_bounds__(256) void stats_kernel(const float* __restrict__ psum,
                                                    const float* __restrict__ psumsq,
                                                    const float* __restrict__ h_mean,
                                                    const float* __restrict__ h_var,
                                                    float* __restrict__ nmean,
                                                    float* __restrict__ nvar,
                                                    float* __restrict__ varpart,
                                                    int Bdim, int D) {
  const int d = blockIdx.x * 256 + threadIdx.x;
  float s = 0.f, sq = 0.f;
  #pragma unroll
  for (int c = 0; c < NCHUNK; ++c) {
    s  += psum  [(size_t)c * D + d];
    sq += psumsq[(size_t)c * D + d];
  }
  const float Bf = (float)Bdim;
  const float U  = 10.0f;                        // H_UPPER
  const float hu = s / Bf;
  const float hv = sq / Bf - hu * hu;            // biased variance
  const float hm = h_mean[d];
  const float hva = h_var[d];
  const float dm = hu - hm;
  const float nv = (hva * (U - 1.0f / Bf) + hv + dm * dm / (1.0f + 1.0f / U))
                   / (U + 1.0f - 1.0f / Bf);
  const float nm = (hm * U + hu) / (U + 1.0f);
  nvar[d]  = nv;
  nmean[d] = nm;

  __shared__ float red[256];
  red[threadIdx.x] = nv;
  __syncthreads();
  for (int off = 128; off > 0; off >>= 1) {
    if (threadIdx.x < off) red[threadIdx.x] += red[threadIdx.x + off];
    __syncthreads();
  }
  if (threadIdx.x == 0) varpart[blockIdx.x] = red[0];
}

// ---------------------------------------------------------------------------
// Kernel 3: reduce partials -> scalar mean(new_var)/100, precompute
// inv_denom[d] = 1 / (new_var[d] + scalar) so pass 2 needs no divides.
// ---------------------------------------------------------------------------
__global__ __launch_bounds__(256) void denom_kernel(const float* __restrict__ varpart,
                                                    int nparts,
                                                    const float* __restrict__ nvar,
                                                    float* __restrict__ invd, int D) {
  __shared__ float red[256];
  float s = 0.f;
  for (int i = threadIdx.x; i < nparts; i += 256) s += varpart[i];
  red[threadIdx.x] = s;
  __syncthreads();
  for (int off = 128; off > 0; off >>= 1) {
    if (threadIdx.x < off) red[threadIdx.x] += red[threadIdx.x + off];
    __syncthreads();
  }
  const float add = red[0] / (float)D * 0.01f;   // mean(new_var)/100
  for (int d = threadIdx.x; d < D; d += 256)
    invd[d] = 1.0f / (nvar[d] + add);
}

// ---------------------------------------------------------------------------
// Kernel 4: one block per row. Streaming copy h -> out (non-temporal, keeps h
// resident in L2) while tracking the top-1 score; patch winner to new_mean.
// Tie-break: strict '>' per thread (ascending d) + min-index on cross-thread
// ties == JAX argmax first-occurrence semantics.
// ---------------------------------------------------------------------------
__global__ __launch_bounds__(256) void punish_kernel(const float* __restrict__ h,
                                                     const float* __restrict__ nmean,
                                                     const float* __restrict__ invd,
                                                     float* __restrict__ out, int D) {
  const int row = blockIdx.x;
  const int t   = threadIdx.x;
  const size_t base = (size_t)row * D;

  float best = -FLT_MAX;
  int   bidx = 0;
  for (int d = t; d < D; d += 256) {
    const float x    = h[base + d];
    const float diff = x - nmean[d];
    const float sc   = diff * diff * invd[d];
    __builtin_nontemporal_store(x, &out[base + d]);
    if (sc > best) { best = sc; bidx = d; }
  }

  __shared__ float sv[256];
  __shared__ int   si[256];
  sv[t] = best; si[t] = bidx;
  __syncthreads();
  for (int off = 128; off > 0; off >>= 1) {
    if (t < off) {
      const float v2 = sv[t + off];
      const int   i2 = si[t + off];
      if (v2 > sv[t] || (v2 == sv[t] && i2 < si[t])) { sv[t] = v2; si[t] = i2; }
    }
    __syncthreads();
  }

  __threadfence();       // order each thread's NT stores before the patch
  __syncthreads();
  if (t == 0) {
    const int w = si[0];
    out[base + w] = nmean[w];   // winner -> running mean
  }
}

// ---------------------------------------------------------------------------
extern "C" void kernel_launch(void* const* d_in, const int* in_sizes, int n_in,
                              void* d_out, int out_size, void* d_ws, size_t ws_size,
                              hipStream_t stream) {
  const float* h      = (const float*)d_in[0];
  const float* h_mean = (const float*)d_in[1];
  const float* h_var  = (const float*)d_in[2];
  float*       out    = (float*)d_out;

  const int D    = in_sizes[1];            // 16384
  const int Bdim = in_sizes[0] / D;        // 2048

  // Workspace layout (all float): psum[8][D] | psumsq[8][D] | nmean[D] |
  // nvar[D] | invd[D] | varpart[D/256]   (~1.2 MiB total)
  float* psum    = (float*)d_ws;
  float* psumsq  = psum   + (size_t)NCHUNK * D;
  float* nmean   = psumsq + (size_t)NCHUNK * D;
  float* nvar    = nmean  + D;
  float* invd    = nvar   + D;
  float* varpart = invd   + D;

  dim3 g1(D / 128, NCHUNK);
  colstat_wmma<<<g1, 256, 0, stream>>>(h, psum, psumsq, Bdim, D);

  const int nblk2 = D / 256;
  stats_kernel<<<nblk2, 256, 0, stream>>>(psum, psumsq, h_mean, h_var,
                                          nmean, nvar, varpart, Bdim, D);

  denom_kernel<<<1, 256, 0, stream>>>(varpart, nblk2, nvar, invd, D);

  punish_kernel<<<Bdim, 256, 0, stream>>>(h, nmean, invd, out, D);
}